// GraphConv_48533130445595
// MI455X (gfx1250) — compile-verified
//
#include <hip/hip_runtime.h>

#define CH 64  // C_IN == C_OUT == 64

typedef __attribute__((ext_vector_type(2))) float v2f;
typedef __attribute__((ext_vector_type(8))) float v8f;

// ---------------- Kernel 1: zero the f accumulator in workspace ----------------
__global__ void gc_zero_f(float4* __restrict__ f4, int n4) {
    int i = blockIdx.x * blockDim.x + threadIdx.x;
    if (i < n4) f4[i] = float4{0.f, 0.f, 0.f, 0.f};
}

// ---------------- Kernel 2: edge scatter  f[dst] += w * X[src] ----------------
// 16 threads per edge, each handles 4 contiguous channels (float4 load, 4 atomics).
__global__ __launch_bounds__(256) void gc_scatter(
    const float* __restrict__ X,
    const int*   __restrict__ esrc,
    const int*   __restrict__ edst,
    const float* __restrict__ ew,
    float*       __restrict__ f,
    int E)
{
    int t = blockIdx.x * blockDim.x + threadIdx.x;
    int e = t >> 4;
    if (e >= E) return;
    int c4 = (t & 15) * 4;

    int s = esrc[e];
    int d = edst[e];
    float w = ew[e];

    const float4 x = *(const float4*)(X + s * CH + c4);
    float* fp = f + d * CH + c4;
    atomicAdd(fp + 0, w * x.x);
    atomicAdd(fp + 1, w * x.y);
    atomicAdd(fp + 2, w * x.z);
    atomicAdd(fp + 3, w * x.w);
}

// ---------------- Kernel 3: out = f @ W + b  via V_WMMA_F32_16X16X4_F32 ----------------
// 128 threads = 4 waves per block. Each wave computes one 16(M)x64(N) strip.
// A (16x4 f32): lane 0-15 -> M=lane, holds K=k0,k0+1 ; lane 16-31 -> M=lane-16, holds K=k0+2,k0+3
// B (4x16 f32): lane 0-15 -> N=lane, holds K=k0,k0+1 ; lane 16-31 -> N=lane-16, holds K=k0+2,k0+3
// C/D (16x16 f32): VGPR r, lanes 0-15 -> D[M=r][N=lane] ; lanes 16-31 -> D[M=8+r][N=lane-16]
//
// W is staged into LDS pre-swizzled as pair-rows:
//   Ws2[pr][n] = { W[2*pr][n], W[2*pr+1][n] }   (pr = 0..31, n = 0..63)
// so each lane's B fragment is one aligned ds_load_b64 (no VGPR shuffle movs).
__global__ __launch_bounds__(128) void gc_gemm_wmma(
    const float* __restrict__ f,
    const float* __restrict__ W,     // [64][64] row-major (K x N)
    const float* __restrict__ bias,  // [64]
    float*       __restrict__ out,   // [V][64]
    int V)
{
    __shared__ float2 Ws2[32 * CH];  // 16 KB, pair-row-major

    int tid = threadIdx.x;
    // Cooperative swizzled stage: 2048 float2, 16 per thread.
#pragma unroll
    for (int i = 0; i < 16; ++i) {
        int idx = tid + 128 * i;          // 0..2047
        int pr  = idx >> 6;               // pair-row index (K/2)
        int n   = idx & 63;               // column
        Ws2[idx] = float2{ W[(2 * pr) * CH + n], W[(2 * pr + 1) * CH + n] };
    }
    __syncthreads();

    int wave = tid >> 5;
    int lane = tid & 31;
    int strip = blockIdx.x * 4 + wave;          // 16-row strip index
    int nStrips = (V + 15) >> 4;                // V=100000 -> 6250 exact
    if (strip >= nStrips) return;               // wave-uniform branch (EXEC stays full)
    int row0 = strip * 16;

    int lmod = lane & 15;
    int lhi  = lane >> 4;                       // 0 or 1

    v8f acc0 = {}, acc1 = {}, acc2 = {}, acc3 = {};

    // A fragment base: row (row0+lmod), K offset starts at 2*lhi
    const float* fbase = f + (row0 + lmod) * CH + 2 * lhi;
    // B fragment base: pair-row starts at lhi, column lmod
    const float2* bbase = Ws2 + lhi * CH + lmod;

#pragma unroll
    for (int k0 = 0; k0 < CH; k0 += 4) {
        const float2 av = *(const float2*)(fbase + k0);
        v2f a; a.x = av.x; a.y = av.y;

        const float2* bp = bbase + (k0 >> 1) * CH;   // pr = k0/2 + lhi
        float2 f0 = bp[0];
        float2 f1 = bp[16];
        float2 f2 = bp[32];
        float2 f3 = bp[48];
        v2f b0; b0.x = f0.x; b0.y = f0.y;
        v2f b1; b1.x = f1.x; b1.y = f1.y;
        v2f b2; b2.x = f2.x; b2.y = f2.y;
        v2f b3; b3.x = f3.x; b3.y = f3.y;

        acc0 = __builtin_amdgcn_wmma_f32_16x16x4_f32(false, a, false, b0, (short)0, acc0, false, false);
        acc1 = __builtin_amdgcn_wmma_f32_16x16x4_f32(false, a, false, b1, (short)0, acc1, false, false);
        acc2 = __builtin_amdgcn_wmma_f32_16x16x4_f32(false, a, false, b2, (short)0, acc2, false, false);
        acc3 = __builtin_amdgcn_wmma_f32_16x16x4_f32(false, a, false, b3, (short)0, acc3, false, false);
    }

    // Store with bias. Lane (lhi,lmod): VGPR r holds D[M = r + 8*lhi][N = ntile*16 + lmod].
    v8f accs[4] = {acc0, acc1, acc2, acc3};
#pragma unroll
    for (int nt = 0; nt < 4; ++nt) {
        int n = nt * 16 + lmod;
        float bv = bias[n];
#pragma unroll
        for (int r = 0; r < 8; ++r) {
            int m = r + 8 * lhi;
            out[(row0 + m) * CH + n] = accs[nt][r] + bv;
        }
    }
}

// ---------------- Launch ----------------
extern "C" void kernel_launch(void* const* d_in, const int* in_sizes, int n_in,
                              void* d_out, int out_size, void* d_ws, size_t ws_size,
                              hipStream_t stream) {
    const float* X    = (const float*)d_in[0];
    const int*   esrc = (const int*)  d_in[1];
    const int*   edst = (const int*)  d_in[2];
    const float* ew   = (const float*)d_in[3];
    const float* W    = (const float*)d_in[4];
    const float* bias = (const float*)d_in[5];

    const int V = in_sizes[0] / CH;   // 100000
    const int E = in_sizes[1];        // 1250000

    float* f = (float*)d_ws;          // [V][64] fp32 accumulator (25.6 MB)
    float* out = (float*)d_out;

    // 1) zero f
    {
        int n4 = (V * CH) / 4;
        int blocks = (n4 + 255) / 256;
        gc_zero_f<<<blocks, 256, 0, stream>>>((float4*)f, n4);
    }
    // 2) scatter edges
    {
        long long threads = (long long)E * 16;
        int blocks = (int)((threads + 255) / 256);
        gc_scatter<<<blocks, 256, 0, stream>>>(X, esrc, edst, ew, f, E);
    }
    // 3) GEMM + bias via WMMA
    {
        int nStrips = (V + 15) / 16;              // 6250
        int blocks  = (nStrips + 3) / 4;          // 4 waves (strips) per block
        gc_gemm_wmma<<<blocks, 128, 0, stream>>>(f, W, bias, out, V);
    }
}